// LSTM_57329223467189
// MI455X (gfx1250) — compile-verified
//
#include <hip/hip_runtime.h>
#include <hip/hip_bf16.h>

// ---------------------------------------------------------------------------
// LSTM (SEQ=64, IN_DIM=1, H=1024, C=10, B=1024) for gfx1250 (MI455X, wave32).
//
//   - 64 fused step kernels: 4x (H,H)@(H,B) WMMA f32 GEMMs (gates share the h
//     operand) + rank-1 x term + bias + activations + c/h update.
//   - fp32 tensor path: V_WMMA_F32_16X16X4_F32 (reference precision).
//   - Each wave: 16(M) x 64(N) tile, 4 gates -> 16 v8f accumulators (128 VGPRs).
//   - h and c live in FRAGMENT-MAJOR layout (identical to the WMMA C/D VGPR
//     layout):  addr(m,n) = ((m>>4)*(B/16) + (n>>4))*256 + lane(m,n)*8 + (m&7)
//     with lane(m,n) = ((m>>3)&1)*16 + (n&15).  So:
//       * epilogue c/h updates are 2x b128 per 16x16 tile per lane,
//       * the WMMA B fragment (K=kr,kr+1) stays one aligned ds_load_b64,
//       * staging is a straight contiguous async copy.
//   - h tiles staged to LDS with GLOBAL_LOAD_ASYNC_TO_LDS_B128 (ASYNCcnt),
//     double buffered, pipelined with s_wait_asynccnt.
//   - Weights (16 MB fp32) + h (4 MB) are L2-resident (192 MB L2) -> the step
//     kernel is compute-bound on the WMMA pipe, not on HBM (23.3 TB/s idle).
// ---------------------------------------------------------------------------

typedef __attribute__((ext_vector_type(2))) float v2f;
typedef __attribute__((ext_vector_type(8))) float v8f;

#define HDIM 1024
#define BDIM 1024
#define SEQLEN 64
#define NCLS 10

#define TM 64          // block M tile (4 waves * 16)
#define TN 64          // block N tile
#define TK 64          // K staging depth in LDS
#define NTHREADS 128   // 4 waves
#define NSUB 4         // N-subtiles of 16 per wave
#define NTILE (BDIM / 16)   // 64 n-tiles per row of tiles

__device__ __forceinline__ float fast_sigmoid(float v) {
    return 1.0f / (1.0f + __expf(-v));
}
__device__ __forceinline__ float fast_tanh(float v) {
    float e = __expf(-2.0f * v);
    return (1.0f - e) / (1.0f + e);
}

// --- CDNA5 async memory->LDS copy (ASYNCcnt path) --------------------------
__device__ __forceinline__ void async_copy_b128(const void* gptr, unsigned lds_byte_addr) {
    // GLOBAL_LOAD_ASYNC_TO_LDS_B128: VDST = per-lane LDS byte address,
    // VADDR = per-lane 64-bit global address (GV mode).
    asm volatile("global_load_async_to_lds_b128 %0, %1, off"
                 :: "v"(lds_byte_addr), "v"(gptr)
                 : "memory");
}
__device__ __forceinline__ void wait_asynccnt_le8() {
    asm volatile("s_wait_asynccnt 0x8" ::: "memory");
}
__device__ __forceinline__ void wait_asynccnt_0() {
    asm volatile("s_wait_asynccnt 0x0" ::: "memory");
}

__global__ void zero_kernel(float* __restrict__ p, int n) {
    int i = blockIdx.x * blockDim.x + threadIdx.x;
    if (i < n) p[i] = 0.0f;
}

// Stage h tiles (k in [ko,ko+TK), n in [nbase,nbase+TN)) into LDS.
// Fragment-major layout -> 4 contiguous 4 KB chunks (one per 16-k tile row).
// 16 KB total; 128 threads x 8 async b128 transfers each.
__device__ __forceinline__ void stage_tile_async(const float* __restrict__ hfr,
                                                 int ko, int nbase, int tid,
                                                 unsigned lds_base_bytes)
{
    const float* gbase = hfr + ((unsigned)(ko >> 4) * NTILE + (unsigned)(nbase >> 4)) * 256u;
    #pragma unroll
    for (int i = 0; i < 8; ++i) {
        int idx = tid + i * NTHREADS;   // 0..1023 b128 slots
        int mt  = idx >> 8;             // k-tile row within stage (0..3)
        int w16 = idx & 255;            // 16B chunk within 4 KB
        async_copy_b128(gbase + (unsigned)mt * (NTILE * 256u) + (unsigned)w16 * 4u,
                        lds_base_bytes + (unsigned)(mt * 4096 + w16 * 16));
    }
}

__global__ __launch_bounds__(NTHREADS)
void lstm_step_kernel(const float* __restrict__ x, int t,
                      const float* __restrict__ Wgh, const float* __restrict__ Wih,
                      const float* __restrict__ Wfh, const float* __restrict__ Woh,
                      const float* __restrict__ Wgx, const float* __restrict__ Wix,
                      const float* __restrict__ Wfx, const float* __restrict__ Wox,
                      const float* __restrict__ bg,  const float* __restrict__ bi,
                      const float* __restrict__ bf,  const float* __restrict__ bo,
                      const float* __restrict__ hprev, float* __restrict__ hnext,
                      float* __restrict__ cstate)
{
    __shared__ __align__(16) float lh[2][TK * TN];   // 2 x 16 KB double buffer

    const int tid  = threadIdx.x;
    const int lane = tid & 31;
    const int wave = tid >> 5;
    const int l15  = lane & 15;
    const int lhi  = lane >> 4;     // 0 or 1

    const int mbase = blockIdx.x * TM + wave * 16;
    const int nbase = blockIdx.y * TN;

    const unsigned lbase0 = (unsigned)(uintptr_t)&lh[0][0];   // LDS byte addrs
    const unsigned lbase1 = (unsigned)(uintptr_t)&lh[1][0];

    // accumulators: [gate][n-subtile], 16x16 f32 tile each -> 128 VGPRs
    v8f acc[4][NSUB];
    #pragma unroll
    for (int g = 0; g < 4; ++g)
        #pragma unroll
        for (int s = 0; s < NSUB; ++s)
            acc[g][s] = (v8f){};

    const float* W[4] = {Wgh, Wih, Wfh, Woh};
    const int arow = mbase + l15;   // A fragment: M = lane%16

    const int NST = HDIM / TK;      // 16 stages

    // prologue: stage 0
    stage_tile_async(hprev, 0, nbase, tid, lbase0);

    for (int st = 0; st < NST; ++st) {
        // issue next stage into the other buffer, then retire current stage
        if (st + 1 < NST) {
            stage_tile_async(hprev, (st + 1) * TK, nbase, tid,
                             ((st + 1) & 1) ? lbase1 : lbase0);
            wait_asynccnt_le8();    // current stage's 8 transfers done (in-order)
        } else {
            wait_asynccnt_0();
        }
        __syncthreads();            // all waves' staging visible

        const float* bufp = &lh[st & 1][0];

        #pragma unroll 4
        for (int k4 = 0; k4 < TK; k4 += 4) {
            // B fragments (shared across all 4 gates), one ds_load_b64 each.
            // B 4x16 layout: VGPR j, lane L -> K = j + 2*(L/16), N = L%16.
            // Fragment-major LDS keeps (K=kr, K=kr+1) adjacent (e, e+1).
            const int kr = k4 + 2 * lhi;            // even, kr&7 <= 6
            const int foff = (kr >> 4) * 1024
                           + (((kr >> 3) & 1) * 16 + l15) * 8 + (kr & 7);
            v2f bfrag[NSUB];
            #pragma unroll
            for (int s = 0; s < NSUB; ++s)
                bfrag[s] = *(const v2f*)&bufp[foff + s * 256];

            // A fragments from global (L2-resident weights), aligned b64:
            // A 16x4 layout: lane L, VGPR j -> M = L%16, K = 2*(L/16) + j
            const int kk = st * TK + k4 + 2 * lhi;
            #pragma unroll
            for (int g = 0; g < 4; ++g) {
                v2f afrag = *(const v2f*)&W[g][arow * HDIM + kk];
                #pragma unroll
                for (int s = 0; s < NSUB; ++s) {
                    acc[g][s] = __builtin_amdgcn_wmma_f32_16x16x4_f32(
                        false, afrag, false, bfrag[s],
                        (short)0, acc[g][s], false, false);
                }
            }
        }
        __syncthreads();            // done reading before buffer is overwritten
    }

    // ---- epilogue: x-term (rank-1), bias, activations, c/h update ----
    // Per-lane m octet: m = mbase + 8*lhi + e, e = 0..7 (matches C/D layout).
    const int mo = mbase + 8 * lhi;
    float wgx[8], wix[8], wfx[8], wox[8], bgv[8], biv[8], bfv[8], bov[8];
    *(float4*)&wgx[0] = *(const float4*)&Wgx[mo]; *(float4*)&wgx[4] = *(const float4*)&Wgx[mo + 4];
    *(float4*)&wix[0] = *(const float4*)&Wix[mo]; *(float4*)&wix[4] = *(const float4*)&Wix[mo + 4];
    *(float4*)&wfx[0] = *(const float4*)&Wfx[mo]; *(float4*)&wfx[4] = *(const float4*)&Wfx[mo + 4];
    *(float4*)&wox[0] = *(const float4*)&Wox[mo]; *(float4*)&wox[4] = *(const float4*)&Wox[mo + 4];
    *(float4*)&bgv[0] = *(const float4*)&bg[mo];  *(float4*)&bgv[4] = *(const float4*)&bg[mo + 4];
    *(float4*)&biv[0] = *(const float4*)&bi[mo];  *(float4*)&biv[4] = *(const float4*)&bi[mo + 4];
    *(float4*)&bfv[0] = *(const float4*)&bf[mo];  *(float4*)&bfv[4] = *(const float4*)&bf[mo + 4];
    *(float4*)&bov[0] = *(const float4*)&bo[mo];  *(float4*)&bov[4] = *(const float4*)&bo[mo + 4];

    float xv[NSUB];
    #pragma unroll
    for (int s = 0; s < NSUB; ++s)
        xv[s] = x[(nbase + s * 16 + l15) * SEQLEN + t];

    const unsigned mt = (unsigned)(blockIdx.x * 4 + wave);   // 16-row m tile id
    #pragma unroll
    for (int s = 0; s < NSUB; ++s) {
        // fragment-major tile base + this lane's 8-float slot
        const unsigned tb = (mt * NTILE + (unsigned)(nbase >> 4) + s) * 256u
                          + (unsigned)lane * 8u;
        float cc[8], hh[8];
        *(float4*)&cc[0] = *(const float4*)&cstate[tb];
        *(float4*)&cc[4] = *(const float4*)&cstate[tb + 4];
        #pragma unroll
        for (int e = 0; e < 8; ++e) {
            float gp = acc[0][s][e] + wgx[e] * xv[s] + bgv[e];
            float ip = acc[1][s][e] + wix[e] * xv[s] + biv[e];
            float fp = acc[2][s][e] + wfx[e] * xv[s] + bfv[e];
            float op = acc[3][s][e] + wox[e] * xv[s] + bov[e];
            float gg = fast_tanh(gp);
            float ii = fast_sigmoid(ip);
            float ff = fast_sigmoid(fp);
            float oo = fast_sigmoid(op);
            float c2 = gg * ii + cc[e] * ff;
            cc[e] = c2;
            hh[e] = fast_tanh(c2) * oo;
        }
        *(float4*)&cstate[tb]     = *(float4*)&cc[0];
        *(float4*)&cstate[tb + 4] = *(float4*)&cc[4];
        *(float4*)&hnext[tb]      = *(float4*)&hh[0];
        *(float4*)&hnext[tb + 4]  = *(float4*)&hh[4];
    }
}

// p = Wph @ h + bp  (C x B); softmax over axis=1 (batch axis, per reference);
// output transposed: out[b * C + c].  h is fragment-major.
__global__ __launch_bounds__(256)
void classifier_kernel(const float* __restrict__ h, const float* __restrict__ Wph,
                       const float* __restrict__ bp, float* __restrict__ out)
{
    __shared__ float pbuf[BDIM];
    __shared__ float red[256];

    const int c   = blockIdx.x;
    const int tid = threadIdx.x;

    for (int b = tid; b < BDIM; b += 256) {
        float s = bp[c];
        const unsigned bt = (unsigned)(b >> 4);
        const unsigned bl = (unsigned)(b & 15);
        for (int kt = 0; kt < HDIM / 16; ++kt) {
            const float* tile = h + ((unsigned)kt * NTILE + bt) * 256u + bl * 8u;
            #pragma unroll
            for (int hi = 0; hi < 2; ++hi) {
                #pragma unroll
                for (int e = 0; e < 8; e += 2) {
                    float2 hv = *(const float2*)&tile[hi * 128 + e];
                    int k = kt * 16 + hi * 8 + e;
                    s += Wph[c * HDIM + k] * hv.x + Wph[c * HDIM + k + 1] * hv.y;
                }
            }
        }
        pbuf[b] = s;
    }
    __syncthreads();

    float mx = -3.402823466e+38f;
    for (int b = tid; b < BDIM; b += 256) mx = fmaxf(mx, pbuf[b]);
    red[tid] = mx; __syncthreads();
    for (int st = 128; st > 0; st >>= 1) {
        if (tid < st) red[tid] = fmaxf(red[tid], red[tid + st]);
        __syncthreads();
    }
    mx = red[0];
    __syncthreads();

    float sm = 0.0f;
    for (int b = tid; b < BDIM; b += 256) {
        float e = __expf(pbuf[b] - mx);
        pbuf[b] = e;
        sm += e;
    }
    red[tid] = sm; __syncthreads();
    for (int st = 128; st > 0; st >>= 1) {
        if (tid < st) red[tid] += red[tid + st];
        __syncthreads();
    }
    float inv = 1.0f / red[0];
    __syncthreads();

    for (int b = tid; b < BDIM; b += 256)
        out[b * NCLS + c] = pbuf[b] * inv;
}

extern "C" void kernel_launch(void* const* d_in, const int* in_sizes, int n_in,
                              void* d_out, int out_size, void* d_ws, size_t ws_size,
                              hipStream_t stream) {
    const float* x   = (const float*)d_in[0];
    const float* Wgx = (const float*)d_in[1];
    const float* Wgh = (const float*)d_in[2];
    const float* Wix = (const float*)d_in[3];
    const float* Wih = (const float*)d_in[4];
    const float* Wfx = (const float*)d_in[5];
    const float* Wfh = (const float*)d_in[6];
    const float* Wox = (const float*)d_in[7];
    const float* Woh = (const float*)d_in[8];
    const float* Wph = (const float*)d_in[9];
    const float* bg  = (const float*)d_in[10];
    const float* bf  = (const float*)d_in[11];
    const float* bi  = (const float*)d_in[12];
    const float* bo  = (const float*)d_in[13];
    const float* bp  = (const float*)d_in[14];

    // workspace layout: [h0 | c | h1]  (h0 and c zero-initialized together)
    float* h0 = (float*)d_ws;
    float* cs = h0 + HDIM * BDIM;
    float* h1 = cs + HDIM * BDIM;

    zero_kernel<<<(2 * HDIM * BDIM + 255) / 256, 256, 0, stream>>>(h0, 2 * HDIM * BDIM);

    float* hbuf[2] = {h0, h1};
    for (int t = 0; t < SEQLEN; ++t) {
        lstm_step_kernel<<<dim3(HDIM / TM, BDIM / TN), NTHREADS, 0, stream>>>(
            x, t,
            Wgh, Wih, Wfh, Woh,
            Wgx, Wix, Wfx, Wox,
            bg, bi, bf, bo,
            hbuf[t & 1], hbuf[(t + 1) & 1], cs);
    }

    // SEQLEN is even -> final h lives in hbuf[0]
    classifier_kernel<<<NCLS, 256, 0, stream>>>(hbuf[0], Wph, bp, (float*)d_out);
}